// SimpleFPSSampling_68247030333789
// MI455X (gfx1250) — compile-verified
//
#include <hip/hip_runtime.h>

// FPS (farthest point sampling) for B=4, N=131072, C=64, 128 iterations.
// fp32-exact distances computed on the matrix pipe: diag(A * A^T) via
// V_WMMA_F32_16X16X4_F32 with A = 16x4 chunks of (p - centroid).

typedef float v8f __attribute__((ext_vector_type(8)));
typedef float v2f __attribute__((ext_vector_type(2)));

#define FPS_C 64
#define FPS_NP 128        // NUM_POINTS
#define TPB 256           // 8 waves of 32
#define WAVES 8

__global__ void fps_init_kernel(float* __restrict__ dist, size_t nDist,
                                unsigned long long* __restrict__ best,
                                unsigned int* __restrict__ counter,
                                int* __restrict__ farthest,
                                const int* __restrict__ finit, int B) {
    size_t i = (size_t)blockIdx.x * blockDim.x + threadIdx.x;
    if (i < nDist) dist[i] = 1e10f;
    if (i < (size_t)B) {
        best[i] = 0ull;
        counter[i] = 0u;
        farthest[i] = finit[i];
    }
}

__device__ __forceinline__ float pick8(const v8f& a, int j) {
    // a[j] via cndmask chain (avoid scratch from dynamic vector index)
    float r = a[0];
    r = (j == 1) ? a[1] : r;
    r = (j == 2) ? a[2] : r;
    r = (j == 3) ? a[3] : r;
    r = (j == 4) ? a[4] : r;
    r = (j == 5) ? a[5] : r;
    r = (j == 6) ? a[6] : r;
    r = (j == 7) ? a[7] : r;
    return r;
}

__global__ void fps_step_kernel(const float* __restrict__ points,  // B*N*C
                                float* __restrict__ dist,          // B*N
                                unsigned long long* __restrict__ best,
                                unsigned int* __restrict__ counter,
                                int* __restrict__ farthest,
                                float* __restrict__ sampled,       // B*128*C
                                float* __restrict__ centOut,       // B*128
                                int N, int t, int blocksPerBatch,
                                int pointsPerBlock) {
    const int b    = blockIdx.y;
    const int blk  = blockIdx.x;
    const int tid  = threadIdx.x;
    const int lane = tid & 31;
    const int wave = tid >> 5;

    const int f = farthest[b];  // centroid chosen by previous iteration
    const float* __restrict__ bp = points + (size_t)b * N * FPS_C;

    // Record this step's centroid (features + index) once per batch.
    if (blk == 0) {
        if (tid < FPS_C)
            sampled[((size_t)b * FPS_NP + t) * FPS_C + tid] = bp[(size_t)f * FPS_C + tid];
        if (tid == 0)
            centOut[b * FPS_NP + t] = (float)f;
    }

    // Per-lane centroid chunks matching the WMMA A-layout:
    // lane holds dims (4i + koff, 4i + koff + 1), koff = 2*(lane>=16).
    const int koff = (lane >> 4) << 1;
    const float* cptr = bp + (size_t)f * FPS_C + koff;
    v2f cen[16];
#pragma unroll
    for (int i = 0; i < 16; ++i)
        cen[i] = *(const v2f*)(cptr + 4 * i);

    const int ppw      = pointsPerBlock / WAVES;  // multiple of 32
    const int waveBase = blk * pointsPerBlock + wave * ppw;

    float bestD = -1.0f;
    int   bestI = 0;

    for (int tb = waveBase; tb < waveBase + ppw; tb += 32) {
        // group X = points tb..tb+15, group Y = tb+16..tb+31
        const float* pX = bp + (size_t)(tb + (lane & 15)) * FPS_C + koff;
        const float* pY = pX + 16 * FPS_C;
        v8f accX = {};
        v8f accY = {};
#pragma unroll
        for (int i = 0; i < 16; ++i) {
            v2f ax = *(const v2f*)(pX + 4 * i);
            v2f ay = *(const v2f*)(pY + 4 * i);
            v2f qx = ax - cen[i];
            v2f qy = ay - cen[i];
            // D = Q * Q^T + acc ; A-layout registers double as the
            // transposed B-layout, so diag(acc) = sum_k (p-c)^2.
            accX = __builtin_amdgcn_wmma_f32_16x16x4_f32(
                false, qx, false, qx, (short)0, accX, false, false);
            accY = __builtin_amdgcn_wmma_f32_16x16x4_f32(
                false, qy, false, qy, (short)0, accY, false, false);
        }

        // Diagonal lives in lanes 0-7 (rows 0-7, vgpr j=lane) and lanes
        // 24-31 (rows 8-15, vgpr j=lane-24).
        const int j  = lane & 7;
        const float sX = pick8(accX, j);
        const float sY = pick8(accY, j);

        // Redistribute: lane l owns point tb+l. Source lane for row m:
        // m<8 -> lane m ; m>=8 -> lane m+16.
        const int lp = lane & 15;
        const int sl = (lp < 8) ? lp : (lp + 16);
        const int dX = __builtin_amdgcn_ds_bpermute(sl << 2, __builtin_bit_cast(int, sX));
        const int dY = __builtin_amdgcn_ds_bpermute(sl << 2, __builtin_bit_cast(int, sY));
        const float d = __builtin_bit_cast(float, (lane < 16) ? dX : dY);

        const int g = tb + lane;                 // batch-local point index
        float* dp = dist + (size_t)b * N + g;
        const float nd = fminf(*dp, d);
        *dp = nd;
        if (nd > bestD) { bestD = nd; bestI = g; }  // strict > keeps lowest idx
    }

    // Block-level argmax: pack (distBits, ~idx) so ties pick lowest index.
    __shared__ unsigned long long blkBest;
    if (tid == 0) blkBest = 0ull;
    __syncthreads();
    const unsigned long long key =
        ((unsigned long long)__builtin_bit_cast(unsigned int, bestD) << 32) |
        (unsigned int)(~bestI);
    atomicMax(&blkBest, key);
    __syncthreads();

    if (tid == 0) {
        atomicMax(&best[b], blkBest);
        __threadfence();
        const unsigned prev = atomicAdd(&counter[b], 1u);
        if (prev == (unsigned)(blocksPerBatch - 1)) {
            // Last block for this batch: consume + reset for next iteration.
            const unsigned long long k2 = atomicExch(&best[b], 0ull);
            farthest[b] = (int)(~(unsigned int)k2);
            atomicExch(&counter[b], 0u);
            __threadfence();
        }
    }
}

extern "C" void kernel_launch(void* const* d_in, const int* in_sizes, int n_in,
                              void* d_out, int out_size, void* d_ws, size_t ws_size,
                              hipStream_t stream) {
    const float* feat  = (const float*)d_in[0];
    const int*   finit = (const int*)d_in[1];

    const int B = in_sizes[1];                               // 4
    const long long total = (long long)in_sizes[0];          // B*N*C
    const int N = (int)(total / ((long long)B * FPS_C));     // 131072

    // workspace layout
    float* dist = (float*)d_ws;
    size_t distBytes = (size_t)B * N * sizeof(float);
    char* p = (char*)d_ws + ((distBytes + 255) & ~(size_t)255);
    unsigned long long* best = (unsigned long long*)p; p += (size_t)B * 8;
    unsigned int* counter = (unsigned int*)p;          p += (size_t)B * 4;
    int* farthest = (int*)p;

    float* sampled = (float*)d_out;                          // B*128*64
    float* centOut = sampled + (size_t)B * FPS_NP * FPS_C;   // B*128 (as float)

    const size_t nDist = (size_t)B * N;
    const int initBlocks = (int)((nDist + TPB - 1) / TPB);
    fps_init_kernel<<<initBlocks, TPB, 0, stream>>>(dist, nDist, best, counter,
                                                    farthest, finit, B);

    const int blocksPerBatch = 128;
    const int pointsPerBlock = N / blocksPerBatch;           // 1024 (8 waves * 128)
    dim3 grid(blocksPerBatch, B);
    for (int t = 0; t < FPS_NP; ++t) {
        fps_step_kernel<<<grid, TPB, 0, stream>>>(feat, dist, best, counter,
                                                  farthest, sampled, centOut,
                                                  N, t, blocksPerBatch,
                                                  pointsPerBlock);
    }
}